// SALoss_31988916420713
// MI455X (gfx1250) — compile-verified
//
#include <hip/hip_runtime.h>
#include <hip/hip_bf16.h>
#include <math.h>

typedef __attribute__((ext_vector_type(2))) float v2f;
typedef __attribute__((ext_vector_type(8))) float v8f;

#define NCLS   20
#define CPAD   32
#define KDIM   128
#define ALPHA  0.7f
#define BETA   1.5f

// Workspace layout (floats):
//   [0,      4096) : sums   [32][128]  (rows 0..19 used)
//   [4096,   4128) : counts [32]
//   [4128,   4160) : intra  [32]
//   [4160,   8256) : means  [32][128]
//   [8256,   8288) : msq    [32]
#define WS_SUMS   0
#define WS_CNT    4096
#define WS_INTRA  4128
#define WS_MEANS  4160
#define WS_MSQ    8256
#define WS_TOTAL  8288

__device__ inline v8f wmma_f32_16x16x4(v2f a, v2f b, v8f c) {
  return __builtin_amdgcn_wmma_f32_16x16x4_f32(false, a, false, b, (short)0, c,
                                               false, false);
}

// ---------------- Kernel 0: zero accumulators -------------------------------
__global__ void k_zero(float* ws) {
  int i = blockIdx.x * blockDim.x + threadIdx.x;
  if (i < WS_MEANS) ws[i] = 0.0f;  // sums + counts + intra
}

// ---------------- Kernel 1: per-class counts + embedding sums ---------------
// One wave handles one point per iteration: 32 lanes x float4 = full 512B row.
__global__ __launch_bounds__(256) void k_pass1(const float* __restrict__ emb,
                                               const int* __restrict__ labels,
                                               float* __restrict__ ws, int N) {
  __shared__ float lsum[NCLS * KDIM];
  __shared__ float lcnt[NCLS];
  const int tid = threadIdx.x;
  for (int i = tid; i < NCLS * KDIM; i += 256) lsum[i] = 0.0f;
  if (tid < NCLS) lcnt[tid] = 0.0f;
  __syncthreads();

  const int lane = tid & 31;
  const int wv = tid >> 5;
  const int gw = blockIdx.x * 8 + wv;     // wave-uniform
  const int tw = gridDim.x * 8;
  for (int p = gw; p < N; p += tw) {      // p is wave-uniform -> scalar label load
    const int t = labels[p];
    const float4 e = *(const float4*)(emb + (size_t)p * KDIM + lane * 4);
    float* dst = lsum + t * KDIM + lane * 4;
    atomicAdd(dst + 0, e.x);
    atomicAdd(dst + 1, e.y);
    atomicAdd(dst + 2, e.z);
    atomicAdd(dst + 3, e.w);
    if (lane == 0) atomicAdd(&lcnt[t], 1.0f);
  }
  __syncthreads();
  for (int i = tid; i < NCLS * KDIM; i += 256) atomicAdd(&ws[WS_SUMS + i], lsum[i]);
  if (tid < NCLS) atomicAdd(&ws[WS_CNT + tid], lcnt[tid]);
}

// ---------------- Kernel 2: means = sums / max(count,1); msq = ||mean||^2 ---
__global__ __launch_bounds__(256) void k_means(float* __restrict__ ws) {
  const int tid = threadIdx.x;
  for (int i = tid; i < CPAD * KDIM; i += 256) {
    const int c = i >> 7;
    ws[WS_MEANS + i] =
        (c < NCLS) ? ws[WS_SUMS + i] / fmaxf(ws[WS_CNT + c], 1.0f) : 0.0f;
  }
  __syncthreads();
  if (tid < CPAD) {
    float s = 0.0f;
    const float* m = ws + WS_MEANS + tid * KDIM;
    for (int k = 0; k < KDIM; ++k) s += m[k] * m[k];
    ws[WS_MSQ + tid] = s;
  }
}

// ---------------- Kernel 3: WMMA pass — Dots = E @ means^T, intra hinge -----
// Per wave: one 16-point tile. D(16x16) accumulated with V_WMMA_F32_16X16X4_F32,
// 2 class-tiles (classes 0..15 / 16..31-pad), 32 k-steps of depth 4.
__global__ __launch_bounds__(256) void k_intra(const float* __restrict__ emb,
                                               const float* __restrict__ points,
                                               const int* __restrict__ labels,
                                               float* __restrict__ ws, int N) {
  __shared__ float lmeans[CPAD * KDIM];   // 16 KB staged means
  __shared__ float lmsq[CPAD];
  __shared__ float lintra[NCLS];
  __shared__ float ldot[8][16 * CPAD];    // per-wave 16 points x 32 classes
  __shared__ float le2[8][32];            // per-wave ||e||^2 partials

  const int tid = threadIdx.x;
  for (int i = tid; i < CPAD * KDIM; i += 256) lmeans[i] = ws[WS_MEANS + i];
  if (tid < CPAD) lmsq[tid] = ws[WS_MSQ + tid];
  if (tid < NCLS) lintra[tid] = 0.0f;
  __syncthreads();

  const int lane = tid & 31;
  const int wv = tid >> 5;
  const int row = lane & 15;    // A row within tile / B class within tile
  const int h = lane >> 4;      // K sub-offset selector

  const int numTiles = N >> 4;
  const int totW = gridDim.x * 8;
  const int iters = (numTiles + totW - 1) / totW;

  for (int it = 0; it < iters; ++it) {
    const int tile = it * totW + blockIdx.x * 8 + wv;
    const bool active = tile < numTiles;

    if (active) {
      const int pr = tile * 16 + row;                  // this lane's point row
      const float* erow = emb + (size_t)pr * KDIM + 2 * h;
      const float* m0 = lmeans + row * KDIM + 2 * h;          // classes 0..15
      const float* m1 = lmeans + (16 + row) * KDIM + 2 * h;   // classes 16..31

      const int nxt = tile + totW;
      if (nxt < numTiles)
        __builtin_prefetch(emb + (size_t)(nxt * 16 + row) * KDIM + 2 * h, 0, 1);

      v8f acc0 = {0.f, 0.f, 0.f, 0.f, 0.f, 0.f, 0.f, 0.f};
      v8f acc1 = {0.f, 0.f, 0.f, 0.f, 0.f, 0.f, 0.f, 0.f};
      float e2p = 0.0f;

#pragma unroll
      for (int k0 = 0; k0 < KDIM; k0 += 4) {
        const v2f a = *(const v2f*)(erow + k0);    // A: row=lane&15, K=k0+2h{,+1}
        const v2f b0 = *(const v2f*)(m0 + k0);     // B: col=class, same K layout
        const v2f b1 = *(const v2f*)(m1 + k0);
        e2p += a.x * a.x + a.y * a.y;
        acc0 = wmma_f32_16x16x4(a, b0, acc0);
        acc1 = wmma_f32_16x16x4(a, b1, acc1);
      }

      // D layout: VGPR j -> point (j + 8*h), class = lane&15 (+16 for acc1)
#pragma unroll
      for (int j = 0; j < 8; ++j) {
        ldot[wv][(j + 8 * h) * CPAD + row] = acc0[j];
        ldot[wv][(j + 8 * h) * CPAD + 16 + row] = acc1[j];
      }
      le2[wv][lane] = e2p;
    }
    __syncthreads();

    if (active && lane < 16) {
      const int p = tile * 16 + lane;
      const int t = labels[p];
      const float dot = ldot[wv][lane * CPAD + t];
      const float e2 = le2[wv][lane] + le2[wv][16 + lane];
      const float d2 = fmaxf(e2 - 2.0f * dot + lmsq[t], 0.0f);
      const float d = sqrtf(d2);
      const float px = points[(size_t)p * 3 + 0];
      const float py = points[(size_t)p * 3 + 1];
      const float pz = points[(size_t)p * 3 + 2];
      const float g = 1.0f / (1.0f + __expf(-sqrtf(px * px + py * py + pz * pz)));
      const float hinge = fmaxf(d - ALPHA, 0.0f);
      atomicAdd(&lintra[t], g * hinge * hinge);
    }
    __syncthreads();
  }

  if (tid < NCLS) atomicAdd(&ws[WS_INTRA + tid], lintra[tid]);
}

// ---------------- Kernel 4: inter-cluster hinge + combine -------------------
__global__ __launch_bounds__(384) void k_final(float* __restrict__ ws,
                                               float* __restrict__ out) {
  __shared__ float s_inter;
  __shared__ float s_intra;
  const int tid = threadIdx.x;
  if (tid == 0) { s_inter = 0.0f; s_intra = 0.0f; }
  __syncthreads();

  if (tid < (NCLS - 1) * (NCLS - 1)) {
    const int i = 1 + tid / (NCLS - 1);
    const int j = 1 + tid % (NCLS - 1);
    if (i != j) {
      const float* mi = ws + WS_MEANS + i * KDIM;
      const float* mj = ws + WS_MEANS + j * KDIM;
      float sq = 0.0f;
      for (int k = 0; k < KDIM; ++k) {
        const float dd = mi[k] - mj[k];
        sq += dd * dd;
      }
      const float dist = sqrtf(sq);
      const float r = fmaxf(BETA - dist, 0.0f);
      atomicAdd(&s_inter, r * r);
    }
  }
  if (tid >= 1 && tid < NCLS)
    atomicAdd(&s_intra, ws[WS_INTRA + tid] / fmaxf(ws[WS_CNT + tid], 1.0f));
  __syncthreads();

  if (tid == 0)
    out[0] = s_intra / (float)NCLS +
             s_inter / ((float)NCLS * (float)(NCLS - 1));
}

// ---------------------------------------------------------------------------
extern "C" void kernel_launch(void* const* d_in, const int* in_sizes, int n_in,
                              void* d_out, int out_size, void* d_ws, size_t ws_size,
                              hipStream_t stream) {
  const float* points = (const float*)d_in[0];
  const int* labels   = (const int*)d_in[1];
  const float* emb    = (const float*)d_in[2];
  float* out = (float*)d_out;
  float* ws = (float*)d_ws;
  const int N = in_sizes[1];  // 262144 points

  k_zero<<<(WS_MEANS + 255) / 256, 256, 0, stream>>>(ws);
  k_pass1<<<256, 256, 0, stream>>>(emb, labels, ws, N);
  k_means<<<1, 256, 0, stream>>>(ws);
  k_intra<<<256, 256, 0, stream>>>(emb, points, labels, ws, N);
  k_final<<<1, 384, 0, stream>>>(ws, out);
}